// Loss1_54717883351217
// MI455X (gfx1250) — compile-verified
//
#include <hip/hip_runtime.h>
#include <hip/hip_bf16.h>

typedef float v4f __attribute__((ext_vector_type(4)));

#define CE_DESC(a, b)                       \
    {                                       \
        float _hi = fmaxf((a), (b));        \
        float _lo = fminf((a), (b));        \
        (a) = _hi;                          \
        (b) = _lo;                          \
    }

// 9-comparator optimal sorting network for 5 elements, descending order.
__device__ __forceinline__ void sort5_desc(float& t0, float& t1, float& t2,
                                           float& t3, float& t4) {
    CE_DESC(t0, t1); CE_DESC(t3, t4);
    CE_DESC(t2, t4);
    CE_DESC(t2, t3); CE_DESC(t1, t4);
    CE_DESC(t0, t3);
    CE_DESC(t0, t2); CE_DESC(t1, t3);
    CE_DESC(t1, t2);
}

// Insert v into a sorted-descending top-5 (t0 >= t1 >= t2 >= t3 >= t4).
// Outer guard is the common-case single compare; inner chain is branchless.
__device__ __forceinline__ void insert_top5(float v, float& t0, float& t1,
                                            float& t2, float& t3, float& t4) {
    if (v > t4) {
        float a = v, b;
        b = fminf(t0, a); t0 = fmaxf(t0, a); a = b;
        b = fminf(t1, a); t1 = fmaxf(t1, a); a = b;
        b = fminf(t2, a); t2 = fmaxf(t2, a); a = b;
        b = fminf(t3, a); t3 = fmaxf(t3, a); a = b;
        t4 = a;
    }
}

// Zero the label element (if it falls in this vector) and run 4 inserts.
__device__ __forceinline__ void proc4(v4f v, int c0, int yrel, float& t0,
                                      float& t1, float& t2, float& t3,
                                      float& t4) {
    v.x = (c0     == yrel) ? 0.0f : v.x;
    v.y = (c0 + 1 == yrel) ? 0.0f : v.y;
    v.z = (c0 + 2 == yrel) ? 0.0f : v.z;
    v.w = (c0 + 3 == yrel) ? 0.0f : v.w;
    insert_top5(v.x, t0, t1, t2, t3, t4);
    insert_top5(v.y, t0, t1, t2, t3, t4);
    insert_top5(v.z, t0, t1, t2, t3, t4);
    insert_top5(v.w, t0, t1, t2, t3, t4);
}

// One wave32 per row: stream the row once with NT b128 loads (4 in flight
// per wave), keep per-lane top-5, butterfly-merge, emit per-row hinge loss.
__global__ __launch_bounds__(256) void topk_margin_rows_kernel(
    const float* __restrict__ x, const int* __restrict__ y,
    float* __restrict__ row_loss, int B, int C) {
    const int lane = threadIdx.x & 31;
    const int wave = threadIdx.x >> 5;
    const int row  = blockIdx.x * 8 + wave;
    if (row >= B) return;

    const size_t base = (size_t)row * (size_t)C;
    const int    ycol = y[row];

    float t0 = -INFINITY, t1 = -INFINITY, t2 = -INFINITY,
          t3 = -INFINITY, t4 = -INFINITY;

    // --- scalar prologue to reach 16-byte alignment (C is odd) ---
    const int skip = (int)((4u - ((unsigned)base & 3u)) & 3u);
    if (lane < skip) {
        float v = __builtin_nontemporal_load(x + base + lane);
        if (lane == ycol) v = 0.0f;
        insert_top5(v, t0, t1, t2, t3, t4);
    }

    // --- main streaming loop: 4x unrolled 16B NT loads for MLP ---
    const int nvec  = (C - skip) >> 2;
    const int ntail = (C - skip) & 3;
    const v4f* __restrict__ vp = (const v4f*)(x + base + skip);
    const int yrel = ycol - skip;  // label column relative to aligned region

    const int nvec4 = nvec & ~127;  // multiple of 128 (= 32 lanes * 4 vecs)
    int i = lane;
    for (; i < nvec4; i += 128) {
        v4f va = __builtin_nontemporal_load(vp + i);
        v4f vb = __builtin_nontemporal_load(vp + i + 32);
        v4f vc = __builtin_nontemporal_load(vp + i + 64);
        v4f vd = __builtin_nontemporal_load(vp + i + 96);
        proc4(va, i << 2,         yrel, t0, t1, t2, t3, t4);
        proc4(vb, (i + 32) << 2,  yrel, t0, t1, t2, t3, t4);
        proc4(vc, (i + 64) << 2,  yrel, t0, t1, t2, t3, t4);
        proc4(vd, (i + 96) << 2,  yrel, t0, t1, t2, t3, t4);
    }
    for (; i < nvec; i += 32) {
        v4f v = __builtin_nontemporal_load(vp + i);
        proc4(v, i << 2, yrel, t0, t1, t2, t3, t4);
    }

    // --- scalar tail ---
    const int tcol = skip + (nvec << 2) + lane;
    if (lane < ntail) {
        float v = __builtin_nontemporal_load(x + base + tcol);
        if (tcol == ycol) v = 0.0f;
        insert_top5(v, t0, t1, t2, t3, t4);
    }

    // --- wave32 butterfly merge of sorted-desc top-5 lists ---
    // Bitonic split: top-5 of union(a,b) = { max(a[i], b[4-i]) }, then resort.
    #pragma unroll
    for (int s = 1; s < 32; s <<= 1) {
        float o0 = __shfl_xor(t0, s, 32);
        float o1 = __shfl_xor(t1, s, 32);
        float o2 = __shfl_xor(t2, s, 32);
        float o3 = __shfl_xor(t3, s, 32);
        float o4 = __shfl_xor(t4, s, 32);
        float c0 = fmaxf(t0, o4);
        float c1 = fmaxf(t1, o3);
        float c2 = fmaxf(t2, o2);
        float c3 = fmaxf(t3, o1);
        float c4 = fmaxf(t4, o0);
        t0 = c0; t1 = c1; t2 = c2; t3 = c3; t4 = c4;
        sort5_desc(t0, t1, t2, t3, t4);
    }

    if (lane == 0) {
        const float s_y = x[base + ycol];  // original (un-zeroed) label score
        row_loss[row] = fmaxf(1.0f + t4 - s_y, 0.0f);
    }
}

// Deterministic fixed-order mean over B per-row losses (no float atomics).
__global__ __launch_bounds__(256) void reduce_mean_kernel(
    const float* __restrict__ row_loss, float* __restrict__ out, int B) {
    __shared__ float smem[8];
    float s = 0.0f;
    for (int i = threadIdx.x; i < B; i += 256) s += row_loss[i];
    #pragma unroll
    for (int m = 1; m < 32; m <<= 1) s += __shfl_xor(s, m, 32);
    if ((threadIdx.x & 31) == 0) smem[threadIdx.x >> 5] = s;
    __syncthreads();
    if (threadIdx.x == 0) {
        float tot = 0.0f;
        #pragma unroll
        for (int w = 0; w < 8; ++w) tot += smem[w];
        out[0] = tot / (float)B;
    }
}

extern "C" void kernel_launch(void* const* d_in, const int* in_sizes, int n_in,
                              void* d_out, int out_size, void* d_ws,
                              size_t ws_size, hipStream_t stream) {
    const float* x = (const float*)d_in[0];
    const int*   y = (const int*)d_in[1];  // jax default x64-off -> int32
    const int B = in_sizes[1];             // 4096
    const int C = in_sizes[0] / B;         // 50257

    float* row_loss = (float*)d_ws;        // B floats of scratch
    float* out      = (float*)d_out;

    topk_margin_rows_kernel<<<(B + 7) / 8, 256, 0, stream>>>(x, y, row_loss, B, C);
    reduce_mean_kernel<<<1, 256, 0, stream>>>(row_loss, out, B);
}